// PostProcess_29480655519784
// MI455X (gfx1250) — compile-verified
//
#include <hip/hip_runtime.h>
#include <hip/hip_bf16.h>

// ---------------- constants from the reference ----------------
#define NEGV        (-1000000000.0f)
#define SCORE_T     (0.5f)
#define IOU_T       (0.45f)
#define AREA_T      (0.0005f)
#define INV_SZ      (1.0f / 416.0f)
#define MAX_BOXES   10
#define NCLS        80
#define NFEAT       85

// decode: LDS-staged tiles of 64 boxes, 64-thread blocks (2 wave32 waves)
#define TILE_BOXES  64
#define TILE_DW     (TILE_BOXES * NFEAT)     // 5440 dwords = 21,760 B per tile
#define TPB_D       64
#define NBLK_D      2048

// suppress: 256-thread blocks, fixed grid
#define TPB_S       256
#define NBLK_S      640
#define TOTAL_S     (TPB_S * NBLK_S)

#define TPB_R       256                      // select/reduce kernel

// Use the Tensor Data Mover on gfx1250 device compiles when the builtin exists.
#if defined(__HIP_DEVICE_COMPILE__) && defined(__gfx1250__) && \
    __has_builtin(__builtin_amdgcn_tensor_load_to_lds)
#define USE_TDM 1
#else
#define USE_TDM 0
#endif

// true clang vector type (usable with __builtin_nontemporal_load, unlike HIP float4)
typedef float v4f __attribute__((ext_vector_type(4)));

__device__ __forceinline__ float clamp01(float x) {
    return fminf(fmaxf(x, 0.0f), 1.0f);
}

__device__ __forceinline__ unsigned lds_byte_addr(const volatile void* p) {
    // flat shared address: low 32 bits are the LDS byte offset (ISA 10.2 aperture map)
    return (unsigned)(unsigned long long)(uintptr_t)p;
}

// block-wide (max value, min index on tie) reduction through LDS.
template <int T>
__device__ __forceinline__ void block_argmax(float v, int idx, float* sV, int* sI) {
    const int t = (int)threadIdx.x;
    sV[t] = v;
    sI[t] = idx;
    __syncthreads();
    for (int s = T >> 1; s > 0; s >>= 1) {
        if (t < s) {
            float ov = sV[t + s];
            int   oi = sI[t + s];
            float cv = sV[t];
            int   ci = sI[t];
            if (ov > cv || (ov == cv && oi < ci)) { sV[t] = ov; sI[t] = oi; }
        }
        __syncthreads();
    }
}

#if USE_TDM
typedef unsigned int v4u __attribute__((ext_vector_type(4)));
typedef int          v4i __attribute__((ext_vector_type(4)));
typedef int          v8i __attribute__((ext_vector_type(8)));

// 1-D TDM copy: ndw dwords from gsrc (global) to lds_byte (LDS). D# per ISA 8.3/8.4:
// g0: count=1 | lds_addr | global_addr[56:0] | type=2
// g1: data_size=4B, tensor_dim0=ndw, tensor_dim1=1, tile_dim0=ndw, tile_dim1=1
__device__ __forceinline__ void tdm_load_tile(const float* gsrc, unsigned lds_byte, int ndw) {
    unsigned long long ga = (unsigned long long)(uintptr_t)gsrc;
    v4u g0;
    g0.x = 1u;                                                  // count=1, user descriptor
    g0.y = lds_byte;                                            // lds_addr (bytes)
    g0.z = (unsigned)ga;                                        // global_addr[31:0]
    g0.w = (unsigned)((ga >> 32) & 0x01FFFFFFu) | (2u << 30);   // addr[56:32] | type=2
    v8i g1;
    g1[0] = (2 << 16);                                          // wg_mask=0, data_size=4B
    g1[1] = (int)(((unsigned)ndw & 0xFFFFu) << 16);             // tensor_dim0[15:0]
    g1[2] = (int)(((unsigned)ndw >> 16) | (1u << 16));          // tensor_dim0[31:16] | tensor_dim1=1
    g1[3] = (int)(((unsigned)ndw & 0xFFFFu) << 16);             // tensor_dim1.hi=0 | tile_dim0=ndw
    g1[4] = 1;                                                  // tile_dim1=1 | tile_dim2=0
    g1[5] = ndw;                                                // tensor_dim0_stride
    g1[6] = 0;
    g1[7] = 0;
    v4i g2 = {0, 0, 0, 0};
    v4i g3 = {0, 0, 0, 0};
#if __clang_major__ >= 23
    v8i g4 = {0, 0, 0, 0, 0, 0, 0, 0};
    __builtin_amdgcn_tensor_load_to_lds(g0, g1, g2, g3, g4, 0);
#else
    __builtin_amdgcn_tensor_load_to_lds(g0, g1, g2, g3, 0);
#endif
}
#endif // USE_TDM

// -------- decode: TDM/coalesced-staged tiles, conflict-free LDS readback -----
__global__ void __launch_bounds__(TPB_D)
decode_kernel(const float* __restrict__ y,
              float4* __restrict__ boxes,
              float*  __restrict__ scores,
              int*    __restrict__ classes,
              float*  __restrict__ work,
              float*  __restrict__ partV,
              int*    __restrict__ partI,
              int N) {
    __shared__ __align__(16) float stage[2][TILE_DW];
    __shared__ float sV[TPB_D];
    __shared__ int   sI[TPB_D];

    if (N < 0) stage[0][0] = 0.0f;   // never executes; keeps 'stage' formally written

    const int  tiles   = (N + TILE_BOXES - 1) / TILE_BOXES;
    const long totalDW = (long)N * NFEAT;

    float bestV = -3.0e38f;
    int   bestI = 0x7fffffff;

    int t   = (int)blockIdx.x;
    int buf = 0;

#if USE_TDM
    const bool issuer = (threadIdx.x < 32);   // wave 0 drives the TDM engine
    if (issuer && t < tiles) {
        long base = (long)t * TILE_DW;
        long rem  = totalDW - base;
        tdm_load_tile(y + base, lds_byte_addr(&stage[0][0]),
                      (int)(rem < TILE_DW ? rem : TILE_DW));
    }
#endif

    for (; t < tiles; t += (int)gridDim.x) {
        const int nt = t + (int)gridDim.x;

#if USE_TDM
        if (issuer) {
            if (nt < tiles) {
                long nb  = (long)nt * TILE_DW;
                long rem = totalDW - nb;
                tdm_load_tile(y + nb, lds_byte_addr(&stage[buf ^ 1][0]),
                              (int)(rem < TILE_DW ? rem : TILE_DW));
                __builtin_amdgcn_s_wait_tensorcnt(1);   // tile t landed (in-order)
            } else {
                __builtin_amdgcn_s_wait_tensorcnt(0);
            }
            __asm__ __volatile__("" ::: "memory");
        }
#else
        {   // fully coalesced 16B staging (read-once -> non-temporal)
            long base = (long)t * TILE_DW;
            long rem  = totalDW - base;
            int  ndw  = (int)(rem < TILE_DW ? rem : TILE_DW);
            const v4f* s4 = (const v4f*)(y + base);
            v4f* d4 = (v4f*)&stage[buf][0];
            int nv = ndw >> 2;
            for (int k = (int)threadIdx.x; k < nv; k += TPB_D)
                d4[k] = __builtin_nontemporal_load(s4 + k);
            for (int k = (nv << 2) + (int)threadIdx.x; k < ndw; k += TPB_D)
                stage[buf][k] = __builtin_nontemporal_load(y + base + k);
            if (nt < tiles)
                __builtin_prefetch(y + (long)nt * TILE_DW + (long)threadIdx.x * 16, 0, 1);
        }
#endif
        __syncthreads();

        const int i = t * TILE_BOXES + (int)threadIdx.x;
        if (i < N) {
            // stride-85 per lane: 85 coprime with 64 banks -> conflict-free ds loads
            const float* p = &stage[buf][threadIdx.x * NFEAT];

            float bx = p[0], by = p[1], bw = p[2], bh = p[3], cf = p[4];

            float mp  = -3.0e38f;
            int   cls = 0;
#pragma unroll 8
            for (int j = 0; j < NCLS; ++j) {
                float pv = p[5 + j];
                if (pv > mp) { mp = pv; cls = j; }   // strict > keeps first index
            }

            float score = cf * mp;

            float xc = clamp01(bx * INV_SZ);
            float yc = clamp01(by * INV_SZ);
            float w  = clamp01(bw * INV_SZ);
            float h  = clamp01(bh * INV_SZ);

            float x1 = clamp01(xc - 0.5f * w);
            float y1 = clamp01(yc - 0.5f * h);
            float x2 = clamp01(xc + 0.5f * w);
            float y2 = clamp01(yc + 0.5f * h);

            float area = (x2 - x1) * (y2 - y1);
            bool  keep = (area > AREA_T) && (score > SCORE_T);
            float wv   = keep ? score : NEGV;

            boxes[i]   = make_float4(x1, y1, x2, y2);
            scores[i]  = score;
            classes[i] = cls;
            work[i]    = wv;

            if (wv > bestV) { bestV = wv; bestI = i; }  // ascending i => first index
        }
        __syncthreads();
        buf ^= 1;
    }

    block_argmax<TPB_D>(bestV, bestI, sV, sI);
    if (threadIdx.x == 0) {
        partV[blockIdx.x] = sV[0];
        partI[blockIdx.x] = sI[0];
    }
}

// ---------------- select round i: reduce partials, emit outputs --------------
__global__ void __launch_bounds__(TPB_R)
select_kernel(const float* __restrict__ partV,
              const int*   __restrict__ partI,
              const float4* __restrict__ boxes,
              const float* __restrict__ scores,
              const int*   __restrict__ classes,
              float* __restrict__ work,
              float* __restrict__ out,
              float* __restrict__ sel,
              int nPart, int iter) {
    __shared__ float sV[TPB_R];
    __shared__ int   sI[TPB_R];

    float bv = -3.0e38f;
    int   bi = 0x7fffffff;
    for (int k = (int)threadIdx.x; k < nPart; k += TPB_R) {
        float v  = partV[k];
        int   id = partI[k];
        if (v > bv || (v == bv && id < bi)) { bv = v; bi = id; }
    }

    block_argmax<TPB_R>(bv, bi, sV, sI);

    if (threadIdx.x == 0) {
        int   j  = sI[0];
        float wv = sV[0];
        float valid = (wv > SCORE_T) ? 1.0f : 0.0f;

        float4 b = boxes[j];
        // layout: boxes[10][4] | scores[10] | classes[10] | valid[10]  (70 floats)
        out[iter * 4 + 0] = b.x;
        out[iter * 4 + 1] = b.y;
        out[iter * 4 + 2] = b.z;
        out[iter * 4 + 3] = b.w;
        out[40 + iter]    = scores[j];
        out[50 + iter]    = (float)classes[j];
        out[60 + iter]    = valid;

        work[j] = NEGV;                 // w.at[j].set(NEG)

        sel[0] = b.x; sel[1] = b.y; sel[2] = b.z; sel[3] = b.w;
        sel[4] = valid;                 // suppression enable for this round
    }
}

// -------- suppress vs selected box, fused with next round's argmax partials --
__global__ void __launch_bounds__(TPB_S)
suppress_kernel(const float4* __restrict__ boxes,
                float* __restrict__ work,
                const float* __restrict__ sel,
                float* __restrict__ partV,
                int*   __restrict__ partI,
                int N) {
    __shared__ float sV[TPB_S];
    __shared__ int   sI[TPB_S];

    const float rx1 = sel[0], ry1 = sel[1], rx2 = sel[2], ry2 = sel[3];
    const bool  v   = sel[4] > 0.5f;
    const float aRef = (rx2 - rx1) * (ry2 - ry1);

    const int t0 = (int)(blockIdx.x * TPB_S + threadIdx.x);
    float bestV = -3.0e38f;
    int   bestI = 0x7fffffff;

    for (int i = t0; i < N; i += TOTAL_S) {
        if (i + TOTAL_S < N)
            __builtin_prefetch(boxes + (i + TOTAL_S), 0, 1);
        float w = work[i];
        if (v && w != NEGV) {           // already-NEG entries stay NEG either way
            float4 b = boxes[i];
            float ix1 = fmaxf(rx1, b.x);
            float iy1 = fmaxf(ry1, b.y);
            float ix2 = fminf(rx2, b.z);
            float iy2 = fminf(ry2, b.w);
            float inter = fmaxf(ix2 - ix1, 0.0f) * fmaxf(iy2 - iy1, 0.0f);
            float a   = (b.z - b.x) * (b.w - b.y);
            float iou = inter / (aRef + a - inter + 1e-10f);
            if (iou > IOU_T) {
                w = NEGV;
                work[i] = w;
            }
        }
        if (w > bestV) { bestV = w; bestI = i; }
    }

    block_argmax<TPB_S>(bestV, bestI, sV, sI);
    if (threadIdx.x == 0) {
        partV[blockIdx.x] = sV[0];
        partI[blockIdx.x] = sI[0];
    }
}

// ----------------------------- host launcher ---------------------------------
extern "C" void kernel_launch(void* const* d_in, const int* in_sizes, int n_in,
                              void* d_out, int out_size, void* d_ws, size_t ws_size,
                              hipStream_t stream) {
    const float* y = (const float*)d_in[0];
    const int N = in_sizes[0] / NFEAT;          // 64 * 10647 = 681,408
    float* out = (float*)d_out;

    // workspace carve-up (~19.2 MB)
    char* ws = (char*)d_ws;
    size_t off = 0;
    float4* boxes   = (float4*)(ws + off); off += (size_t)N * sizeof(float4);
    float*  scores  = (float*) (ws + off); off += (size_t)N * sizeof(float);
    int*    classes = (int*)   (ws + off); off += (size_t)N * sizeof(int);
    float*  work    = (float*) (ws + off); off += (size_t)N * sizeof(float);
    float*  partDV  = (float*) (ws + off); off += (size_t)NBLK_D * sizeof(float);
    int*    partDI  = (int*)   (ws + off); off += (size_t)NBLK_D * sizeof(int);
    float*  partSV  = (float*) (ws + off); off += (size_t)NBLK_S * sizeof(float);
    int*    partSI  = (int*)   (ws + off); off += (size_t)NBLK_S * sizeof(int);
    float*  sel     = (float*) (ws + off); off += 8 * sizeof(float);

    decode_kernel<<<NBLK_D, TPB_D, 0, stream>>>(y, boxes, scores, classes, work,
                                                partDV, partDI, N);

    for (int it = 0; it < MAX_BOXES; ++it) {
        const float* pv = (it == 0) ? partDV : partSV;
        const int*   pi = (it == 0) ? partDI : partSI;
        const int    np = (it == 0) ? NBLK_D : NBLK_S;
        select_kernel<<<1, TPB_R, 0, stream>>>(pv, pi, boxes, scores, classes,
                                               work, out, sel, np, it);
        if (it < MAX_BOXES - 1) {
            suppress_kernel<<<NBLK_S, TPB_S, 0, stream>>>(boxes, work, sel,
                                                          partSV, partSI, N);
        }
    }
}